// RNN_33079838114680
// MI455X (gfx1250) — compile-verified
//
#include <hip/hip_runtime.h>

// GRU forward, B=256, T=2048, H=50, fp32, returns final hidden state (B,H).
// One workgroup per 16 batch rows; persistent scan over T with fused
// [x_t | h] x [W_ih; W_hh] matmul done via V_WMMA_F32_16X16X4_F32.

typedef __attribute__((ext_vector_type(2))) float v2f;
typedef __attribute__((ext_vector_type(8))) float v8f;

#define B_TOT   256
#define T_LEN   2048
#define HDIM    50
#define KDIM    100            // fused reduction: 50 (x) + 50 (h)
#define KSTEPS  25             // KDIM / 4 per V_WMMA_F32_16X16X4_F32
#define BM      16             // batch rows per block (WMMA M)
#define NW      16             // waves per block: 4 col-groups x 4 tiles
#define NTHREADS (NW * 32)
#define A_STRIDE 108           // A-panel row stride (floats); 16 rows -> 16 distinct banks
#define G_STRIDE 260           // gate-buffer row stride (256 cols + pad)

__device__ __forceinline__ float fsigmoid(float v) {
    return 1.0f / (1.0f + __expf(-v));
}
__device__ __forceinline__ float ftanh_fast(float v) {
    v = fminf(15.0f, fmaxf(-15.0f, v));
    float e = __expf(2.0f * v);
    return (e - 1.0f) / (e + 1.0f);
}

__global__ __launch_bounds__(NTHREADS, 1)
void gru_scan_kernel(const float* __restrict__ x,    // (B, T, H)
                     const float* __restrict__ Wih,  // (3H, H)
                     const float* __restrict__ Whh,  // (3H, H)
                     const float* __restrict__ bih,  // (3H)
                     const float* __restrict__ bhh,  // (3H)
                     float* __restrict__ out)        // (B, H)
{
    // LDS: double-buffered A panel [x(0..49) | h(50..99) | pad], plus gate buffer.
    __shared__ float Apan[2][BM][A_STRIDE];   // 2*16*108*4 = 13824 B
    __shared__ float Gbuf[BM][G_STRIDE];      // 16*260*4   = 16640 B

    const int tid   = (int)threadIdx.x;
    const int lane  = tid & 31;
    const int wave  = tid >> 5;
    const int b0    = (int)blockIdx.x * BM;

    // ---- loop-invariant WMMA B-operand (fused weights) in registers ----
    // wave owns output cols [16*wave, 16*wave+16); col group g = wave>>2:
    //   g0: r (Wih_r | Whh_r), g1: z (Wih_z | Whh_z),
    //   g2: gx_n (Wih_n | 0),  g3: gh_n (0 | Whh_n)
    const int g    = wave >> 2;
    const int jg   = ((wave & 3) << 4) + (lane & 15);   // gate index within group
    const bool jok = (jg < HDIM);
    const int koff = (lane >> 4) << 1;                  // 0 (lanes 0-15) or 2 (lanes 16-31)

    float bw[KSTEPS][2];
#pragma unroll
    for (int k = 0; k < KSTEPS; ++k) {
#pragma unroll
        for (int v = 0; v < 2; ++v) {
            const int kk = 4 * k + koff + v;            // 0..99
            float wv = 0.0f;
            if (jok) {
                if (g < 2) {
                    const int row = g * HDIM + jg;      // r rows 0-49, z rows 50-99
                    wv = (kk < HDIM) ? Wih[row * HDIM + kk]
                                     : Whh[row * HDIM + (kk - HDIM)];
                } else if (g == 2) {
                    if (kk < HDIM) wv = Wih[(2 * HDIM + jg) * HDIM + kk];
                } else {
                    if (kk >= HDIM) wv = Whh[(2 * HDIM + jg) * HDIM + (kk - HDIM)];
                }
            }
            bw[k][v] = wv;
        }
    }

    // loop-invariant bias for this lane's output column
    float bias = 0.0f;
    if (jok) {
        if (g < 2)       bias = bih[g * HDIM + jg] + bhh[g * HDIM + jg];
        else if (g == 2) bias = bih[2 * HDIM + jg];
        else             bias = bhh[2 * HDIM + jg];
    }

    // loop-invariant indices: each thread owns elements tid and tid+512 of the
    // 800 = BM*H (m,j) pairs, for x staging, gate math, and output.
    const int m0 = tid / HDIM, j0 = tid - m0 * HDIM;          // tid < 512 < 800
    const int i1 = tid + NTHREADS;
    const bool has1 = (i1 < BM * HDIM);
    const int m1 = has1 ? (i1 / HDIM) : 0;
    const int j1 = has1 ? (i1 - m1 * HDIM) : 0;
    const int cn    = (wave << 4) + (lane & 15);              // C-tile column 0..255
    const int mbase = (lane >> 4) << 3;                       // C rows: 0..7 or 8..15

    // ---- init: zero panel 0 (h0 = 0), load x_0 ----
    for (int i = tid; i < BM * A_STRIDE; i += NTHREADS)
        (&Apan[0][0][0])[i] = 0.0f;
    __syncthreads();
    Apan[0][m0][j0] = x[((size_t)(b0 + m0) * T_LEN) * HDIM + j0];
    if (has1)
        Apan[0][m1][j1] = x[((size_t)(b0 + m1) * T_LEN) * HDIM + j1];
    __syncthreads();

    // ---- sequential scan over T ----
    for (int t = 0; t < T_LEN; ++t) {
        const int cur = t & 1, nxt = cur ^ 1;

        // prefetch x_{t+1} into registers (overlaps with WMMA chain)
        float pf0 = 0.0f, pf1 = 0.0f;
        if (t + 1 < T_LEN) {
            pf0 = x[((size_t)(b0 + m0) * T_LEN + (t + 1)) * HDIM + j0];
            if (has1)
                pf1 = x[((size_t)(b0 + m1) * T_LEN + (t + 1)) * HDIM + j1];
        }

        // Two interleaved accumulator chains (even k -> c0 w/ bias, odd -> c1):
        // halves the serial WMMA dependency depth on the critical path.
        v8f c0, c1;
#pragma unroll
        for (int r = 0; r < 8; ++r) { c0[r] = bias; c1[r] = 0.0f; }

        const float* Arow = &Apan[cur][lane & 15][0];
#pragma unroll
        for (int k = 0; k < KSTEPS; k += 2) {
            v2f a0 = *(const v2f*)(Arow + 4 * k + koff);       // 8B-aligned ds load
            v2f b0v; b0v[0] = bw[k][0]; b0v[1] = bw[k][1];
            c0 = __builtin_amdgcn_wmma_f32_16x16x4_f32(
                     false, a0, false, b0v, (short)0, c0, false, false);
            if (k + 1 < KSTEPS) {
                v2f a1 = *(const v2f*)(Arow + 4 * (k + 1) + koff);
                v2f b1v; b1v[0] = bw[k + 1][0]; b1v[1] = bw[k + 1][1];
                c1 = __builtin_amdgcn_wmma_f32_16x16x4_f32(
                         false, a1, false, b1v, (short)0, c1, false, false);
            }
        }

        // spill pre-activations to gate buffer; stage x_{t+1} into next panel
#pragma unroll
        for (int r = 0; r < 8; ++r)
            Gbuf[mbase + r][cn] = c0[r] + c1[r];
        if (t + 1 < T_LEN) {
            Apan[nxt][m0][j0] = pf0;
            if (has1) Apan[nxt][m1][j1] = pf1;
        }
        __syncthreads();

        // gate math: 800 elements; this thread handles (m0,j0) and (m1,j1)
        {
            const float ar  = Gbuf[m0][j0];
            const float az  = Gbuf[m0][64 + j0];
            const float gxn = Gbuf[m0][128 + j0];
            const float ghn = Gbuf[m0][192 + j0];
            const float r   = fsigmoid(ar);
            const float zz  = fsigmoid(az);
            const float n   = ftanh_fast(gxn + r * ghn);
            const float hp  = Apan[cur][m0][HDIM + j0];
            Apan[nxt][m0][HDIM + j0] = (1.0f - zz) * n + zz * hp;
        }
        if (has1) {
            const float ar  = Gbuf[m1][j1];
            const float az  = Gbuf[m1][64 + j1];
            const float gxn = Gbuf[m1][128 + j1];
            const float ghn = Gbuf[m1][192 + j1];
            const float r   = fsigmoid(ar);
            const float zz  = fsigmoid(az);
            const float n   = ftanh_fast(gxn + r * ghn);
            const float hp  = Apan[cur][m1][HDIM + j1];
            Apan[nxt][m1][HDIM + j1] = (1.0f - zz) * n + zz * hp;
        }
        __syncthreads();
    }

    // final hidden state lives in panel (T_LEN & 1)
    {
        const int par = T_LEN & 1;
        out[(b0 + m0) * HDIM + j0] = Apan[par][m0][HDIM + j0];
        if (has1)
            out[(b0 + m1) * HDIM + j1] = Apan[par][m1][HDIM + j1];
    }
}

extern "C" void kernel_launch(void* const* d_in, const int* in_sizes, int n_in,
                              void* d_out, int out_size, void* d_ws, size_t ws_size,
                              hipStream_t stream) {
    (void)in_sizes; (void)n_in; (void)d_ws; (void)ws_size; (void)out_size;
    const float* x   = (const float*)d_in[0];
    const float* Wih = (const float*)d_in[1];
    const float* Whh = (const float*)d_in[2];
    const float* bih = (const float*)d_in[3];
    const float* bhh = (const float*)d_in[4];
    // d_in[5] = z (scalar), unused
    float* out = (float*)d_out;

    gru_scan_kernel<<<dim3(B_TOT / BM), dim3(NTHREADS), 0, stream>>>(
        x, Wih, Whh, bih, bhh, out);
}